// AttentionBlock_35321811042708
// MI455X (gfx1250) — compile-verified
//
#include <hip/hip_runtime.h>
#include <math.h>

// ---------------------------------------------------------------------------
// AttentionBlock for MI455X (gfx1250, wave32) using V_WMMA_F32_16X16X4_F32.
// B=4, S=2048, H=1024, NH=16, HD=64. All fp32 (matches reference numerics).
// B-operand tiles are stored k-pair-interleaved in LDS so every WMMA
// B-fragment is a single contiguous 8B-aligned ds_load_b64 (no repack movs).
// ---------------------------------------------------------------------------

typedef __attribute__((ext_vector_type(2))) float v2f;
typedef __attribute__((ext_vector_type(8))) float v8f;

#define BATCH  4
#define SEQ    2048
#define HDIM   1024
#define NHEADS 16
#define HEADD  64
#define MTOT   (BATCH * SEQ)          // 8192 rows
#define LSTR   68                     // row-major A/K tiles: 64+4 pad
#define WPSTR  264                    // GEMM W tile, k-pair interleaved: 128*2+8
#define VPSTR  136                    // attn V tile, k-pair interleaved: 64*2+8

__device__ __forceinline__ v8f wmma_f32(v2f a, v2f b, v8f c) {
    return __builtin_amdgcn_wmma_f32_16x16x4_f32(
        false, a, false, b, (short)0, c, false, false);
}

__device__ __forceinline__ v2f lds_pair(const float* p) {  // 8B-aligned pair
    return *(const v2f*)p;
}

__device__ __forceinline__ float rmax16(float v) {
    v = fmaxf(v, __shfl_xor(v, 1));
    v = fmaxf(v, __shfl_xor(v, 2));
    v = fmaxf(v, __shfl_xor(v, 4));
    v = fmaxf(v, __shfl_xor(v, 8));
    return v;
}
__device__ __forceinline__ float rsum16(float v) {
    v += __shfl_xor(v, 1);
    v += __shfl_xor(v, 2);
    v += __shfl_xor(v, 4);
    v += __shfl_xor(v, 8);
    return v;
}

// ---------------------------------------------------------------------------
// C[M,1024] = A[M,1024] @ W[1024,1024] + bias (+ resid).
// Block = 128 threads (4 waves), tile 64M x 128N; waves split along M,
// each wave: 16M x 128N (8 accumulators, 8x A-frag reuse).
// ---------------------------------------------------------------------------
__global__ __launch_bounds__(128)
void gemm_bias_kernel(const float* __restrict__ A, const float* __restrict__ W,
                      const float* __restrict__ bias, const float* __restrict__ resid,
                      float* __restrict__ C)
{
    __shared__ float As[64 * LSTR];        // 64 M-rows x 64 k, row-major
    __shared__ float Wp[32 * WPSTR];       // 32 k-pairs x 128 cols, interleaved

    const int tid  = threadIdx.x;
    const int wave = tid >> 5;
    const int lane = tid & 31;
    const int hh   = lane >> 4;
    const int l16  = lane & 15;
    const int mblk = blockIdx.y * 64;
    const int nblk = blockIdx.x * 128;

    v8f acc[8];
    #pragma unroll
    for (int nt = 0; nt < 8; ++nt) acc[nt] = 0.0f;

    // A staging: 16 lanes x float4 cover one 64-float row; 8 rows/pass
    const int ar = tid >> 4;               // 0..7
    const int ac = (tid & 15) * 4;
    // W staging: 32 lanes x float4 cover one 128-float row-pair; 4 pairs/pass
    const int wr2 = tid >> 5;              // 0..3
    const int wc  = (tid & 31) * 4;        // 0..124

    const float* Abase = A + (size_t)mblk * HDIM;
    const float* Wbase = W + nblk;

    for (int kc = 0; kc < HDIM; kc += 64) {
        __syncthreads();
        #pragma unroll
        for (int i = 0; i < 8; ++i) {
            const int r = ar + i * 8;
            *(float4*)(&As[r * LSTR + ac]) =
                *(const float4*)(Abase + (size_t)r * HDIM + kc + ac);
        }
        #pragma unroll
        for (int i = 0; i < 8; ++i) {
            const int k2 = wr2 + i * 4;    // 0..31
            const float4 w0 = *(const float4*)(Wbase + (size_t)(kc + 2 * k2) * HDIM + wc);
            const float4 w1 = *(const float4*)(Wbase + (size_t)(kc + 2 * k2 + 1) * HDIM + wc);
            float4 lo, hi;                 // zip rows k, k+1
            lo.x = w0.x; lo.y = w1.x; lo.z = w0.y; lo.w = w1.y;
            hi.x = w0.z; hi.y = w1.z; hi.z = w0.w; hi.w = w1.w;
            *(float4*)(&Wp[k2 * WPSTR + wc * 2])     = lo;
            *(float4*)(&Wp[k2 * WPSTR + wc * 2 + 4]) = hi;
        }
        __syncthreads();

        #pragma unroll
        for (int kk = 0; kk < 64; kk += 4) {
            const v2f a = lds_pair(&As[(wave * 16 + l16) * LSTR + kk + 2 * hh]);
            const float* wp = &Wp[((kk >> 1) + hh) * WPSTR + 2 * l16];
            #pragma unroll
            for (int nt = 0; nt < 8; ++nt) {
                const v2f b = lds_pair(wp + nt * 32);   // rows k,k+1 contiguous
                acc[nt] = wmma_f32(a, b, acc[nt]);
            }
        }
    }

    #pragma unroll
    for (int nt = 0; nt < 8; ++nt) {
        const int col = nblk + nt * 16 + l16;
        const float bval = bias[col];
        #pragma unroll
        for (int j = 0; j < 8; ++j) {
            const int row = mblk + wave * 16 + j + 8 * hh;
            const size_t off = (size_t)row * HDIM + col;
            float v = acc[nt][j] + bval;
            if (resid) v += resid[off];
            C[off] = v;
        }
    }
}

// ---------------------------------------------------------------------------
// Flash attention per (b, head, 64-query tile). 4 waves x 16 queries each.
//   S = (Q*scale) K^T   (WMMA; K tile row-major in LDS: frags are b64 pairs)
//   online softmax      (16-lane shfl row reductions; radial mask folded in P)
//   O += P V            (WMMA; V tile k-pair interleaved in LDS)
// ---------------------------------------------------------------------------
__global__ __launch_bounds__(128)
void attn_kernel(const float* __restrict__ Qm, const float* __restrict__ Km,
                 const float* __restrict__ Vm, const float* __restrict__ rmask,
                 float* __restrict__ O)
{
    __shared__ float Ks[64 * LSTR];            // 64 keys x 64 d, row-major
    __shared__ float Vp[32 * VPSTR];           // 32 key-pairs x 64 d, interleaved
    __shared__ float Ps[4][16 * LSTR];         // per-wave P staging

    const int tid  = threadIdx.x;
    const int wave = tid >> 5;
    const int lane = tid & 31;
    const int hh   = lane >> 4;
    const int l16  = lane & 15;

    const int qt    = blockIdx.x;              // 0..31
    const int head  = blockIdx.y;              // 0..15
    const int b     = blockIdx.z;              // 0..3
    const int qbase = qt * 64 + wave * 16;
    const size_t b0 = (size_t)b * SEQ * HDIM;
    const int hcol  = head * HEADD;

    // Preload Q A-fragments (16 d-steps), softmax scale folded in (1/sqrt(64)).
    v2f qf[16];
    {
        const float* qrow = Qm + b0 + (size_t)(qbase + l16) * HDIM + hcol;
        #pragma unroll
        for (int dd = 0; dd < 16; ++dd) {
            const int c = dd * 4 + 2 * hh;
            qf[dd].x = qrow[c]     * 0.125f;
            qf[dd].y = qrow[c + 1] * 0.125f;
        }
    }

    v8f oacc[4];
    #pragma unroll
    for (int n = 0; n < 4; ++n) oacc[n] = 0.0f;
    float mrow[8], lrow[8];
    #pragma unroll
    for (int j = 0; j < 8; ++j) { mrow[j] = -1e30f; lrow[j] = 0.0f; }

    float* Pw = &Ps[wave][0];
    const int sr = tid >> 4;                   // 0..7
    const int sc = (tid & 15) * 4;             // 0..60

    for (int kt = 0; kt < SEQ; kt += 64) {
        __syncthreads();
        // K tile: row-major (QK^T B-frags are contiguous d-pairs already)
        #pragma unroll
        for (int i = 0; i < 8; ++i) {
            const int r = sr + i * 8;
            *(float4*)(&Ks[r * LSTR + sc]) =
                *(const float4*)(Vm == nullptr ? nullptr
                                               : Km + b0 + (size_t)(kt + r) * HDIM + hcol + sc);
        }
        // V tile: key-pair interleaved (P@V B-frags become contiguous pairs)
        #pragma unroll
        for (int i = 0; i < 4; ++i) {
            const int k2 = sr + i * 8;         // 0..31
            const float4 v0 = *(const float4*)(Vm + b0 + (size_t)(kt + 2 * k2) * HDIM + hcol + sc);
            const float4 v1 = *(const float4*)(Vm + b0 + (size_t)(kt + 2 * k2 + 1) * HDIM + hcol + sc);
            float4 lo, hi;
            lo.x = v0.x; lo.y = v1.x; lo.z = v0.y; lo.w = v1.y;
            hi.x = v0.z; hi.y = v1.z; hi.z = v0.w; hi.w = v1.w;
            *(float4*)(&Vp[k2 * VPSTR + sc * 2])     = lo;
            *(float4*)(&Vp[k2 * VPSTR + sc * 2 + 4]) = hi;
        }
        __syncthreads();

        // ---- S = Q K^T : 16 d-steps x 4 key sub-tiles ----
        v8f sacc[4];
        #pragma unroll
        for (int n = 0; n < 4; ++n) sacc[n] = 0.0f;
        #pragma unroll
        for (int dd = 0; dd < 16; ++dd) {
            const v2f a = qf[dd];
            #pragma unroll
            for (int n = 0; n < 4; ++n) {
                const v2f bf =
                    lds_pair(&Ks[(n * 16 + l16) * LSTR + dd * 4 + 2 * hh]);
                sacc[n] = wmma_f32(a, bf, sacc[n]);
            }
        }

        float rmv[4];
        #pragma unroll
        for (int n = 0; n < 4; ++n)
            rmv[n] = rmask[b * SEQ + kt + n * 16 + l16];

        // ---- online softmax (row j+8*hh in VGPR j, key cols in lanes) ----
        float corr[8];
        #pragma unroll
        for (int j = 0; j < 8; ++j) {
            float t = fmaxf(fmaxf(sacc[0][j], sacc[1][j]),
                            fmaxf(sacc[2][j], sacc[3][j]));
            t = rmax16(t);
            const float mn = fmaxf(mrow[j], t);
            corr[j] = __expf(mrow[j] - mn);
            mrow[j] = mn;
            float rs = 0.0f;
            #pragma unroll
            for (int n = 0; n < 4; ++n) {
                const float p = __expf(sacc[n][j] - mn);
                sacc[n][j] = p * rmv[n];       // mask folded into numerator
                rs += p;                       // Z accumulates UNMASKED exp sum
            }
            rs = rsum16(rs);
            lrow[j] = lrow[j] * corr[j] + rs;
        }

        // rescale O, spill P to LDS row-major (C-layout -> A-layout relay)
        #pragma unroll
        for (int n = 0; n < 4; ++n) {
            #pragma unroll
            for (int j = 0; j < 8; ++j) {
                oacc[n][j] *= corr[j];
                Pw[(j + 8 * hh) * LSTR + n * 16 + l16] = sacc[n][j];
            }
        }
        asm volatile("s_wait_dscnt 0" ::: "memory");

        // ---- O += P V : 16 key-steps x 4 d sub-tiles ----
        #pragma unroll
        for (int ks = 0; ks < 16; ++ks) {
            const v2f a = lds_pair(&Pw[l16 * LSTR + ks * 4 + 2 * hh]);
            const float* vp = &Vp[(2 * ks + hh) * VPSTR + 2 * l16];
            #pragma unroll
            for (int n = 0; n < 4; ++n) {
                const v2f bf = lds_pair(vp + n * 32);  // key rows contiguous
                oacc[n] = wmma_f32(a, bf, oacc[n]);
            }
        }
    }

    // normalize by Z and store [b, q, head*64 + d]
    #pragma unroll
    for (int j = 0; j < 8; ++j) {
        const float inv = 1.0f / lrow[j];
        const size_t off = b0 + (size_t)(qbase + j + 8 * hh) * HDIM + hcol;
        #pragma unroll
        for (int n = 0; n < 4; ++n)
            O[off + n * 16 + l16] = oacc[n][j] * inv;
    }
}

// ---------------------------------------------------------------------------
extern "C" void kernel_launch(void* const* d_in, const int* in_sizes, int n_in,
                              void* d_out, int out_size, void* d_ws, size_t ws_size,
                              hipStream_t stream) {
    const float* inputs = (const float*)d_in[0];
    const float* rmask  = (const float*)d_in[1];
    const float* Wq = (const float*)d_in[2];
    const float* bq = (const float*)d_in[3];
    const float* Wk = (const float*)d_in[4];
    const float* bk = (const float*)d_in[5];
    const float* Wv = (const float*)d_in[6];
    const float* bv = (const float*)d_in[7];
    const float* Wo = (const float*)d_in[8];
    const float* bo = (const float*)d_in[9];
    float* out = (float*)d_out;

    const size_t MSH = (size_t)MTOT * HDIM;     // 8192*1024 elements
    float* Qbuf = (float*)d_ws;
    float* Kbuf = Qbuf + MSH;
    float* Vbuf = Qbuf + 2 * MSH;
    float* Abuf = Qbuf;  // attention output aliases Q (disjoint per-block
                         // read-before-write regions; safe)

    dim3 gblock(128);
    dim3 ggrid(HDIM / 128, MTOT / 64);          // (8, 128)

    gemm_bias_kernel<<<ggrid, gblock, 0, stream>>>(inputs, Wq, bq, nullptr, Qbuf);
    gemm_bias_kernel<<<ggrid, gblock, 0, stream>>>(inputs, Wk, bk, nullptr, Kbuf);
    gemm_bias_kernel<<<ggrid, gblock, 0, stream>>>(inputs, Wv, bv, nullptr, Vbuf);

    dim3 agrid(SEQ / 64, NHEADS, BATCH);        // (32, 16, 4)
    attn_kernel<<<agrid, gblock, 0, stream>>>(Qbuf, Kbuf, Vbuf, rmask, Abuf);

    gemm_bias_kernel<<<ggrid, gblock, 0, stream>>>(Abuf, Wo, bo, inputs, out);
}